// ProteinInteractionPredictor_111669149893
// MI455X (gfx1250) — compile-verified
//
#include <hip/hip_runtime.h>
#include <math.h>

typedef __attribute__((ext_vector_type(2))) float v2f;
typedef __attribute__((ext_vector_type(8))) float v8f;

// ---------------------------------------------------------------------------
// GEMM: out[:, col_off:col_off+128] = act(A[N x K] @ W[K x 128] + bias)
// One block = 16 rows x 128 cols. 4 waves; each wave owns two 16x16 col tiles.
// f32 WMMA 16x16x4, fp32 accumulate (matches fp32 reference numerics).
// A-layout (ISA 7.12.2): lane m = lane%16, VGPR r = K (r + 2*(lane/16)).
// B/C/D : lane n = lane%16, VGPR r = row r + 8*(lane/16) (B: r + 2*half).
// ---------------------------------------------------------------------------
__global__ __launch_bounds__(128) void gemm_bias_act_wmma(
    const float* __restrict__ A, const float* __restrict__ W,
    const float* __restrict__ bias, float* __restrict__ out,
    int N, int K, int ldo, int col_off, int do_relu)
{
    const int row0 = blockIdx.x << 4;
    const int wave = threadIdx.x >> 5;
    const int lane = threadIdx.x & 31;
    const int n    = lane & 15;      // A row-in-tile / B,C,D column-in-tile
    const int half = lane >> 4;
    const int col0 = wave << 5;      // this wave: cols [col0, col0+32)

    // Clamp A row so EXEC remains all-1s through WMMA (ISA requirement).
    int ar = row0 + n; if (ar >= N) ar = N - 1;
    const float* arow = A + (size_t)ar * K + 2 * half;

    v8f c0 = {}; v8f c1 = {};
    #pragma unroll 4
    for (int k = 0; k < K; k += 4) {
        v2f a; a.x = arow[k]; a.y = arow[k + 1];
        const float* w0 = W + (size_t)(k + 2 * half) * 128;
        v2f b0, b1;
        b0.x = w0[col0 + n];        b0.y = w0[128 + col0 + n];
        b1.x = w0[col0 + 16 + n];   b1.y = w0[128 + col0 + 16 + n];
        c0 = __builtin_amdgcn_wmma_f32_16x16x4_f32(false, a, false, b0,
                                                   (short)0, c0, false, false);
        c1 = __builtin_amdgcn_wmma_f32_16x16x4_f32(false, a, false, b1,
                                                   (short)0, c1, false, false);
    }

    const float bias0 = bias ? bias[col0 + n]      : 0.0f;
    const float bias1 = bias ? bias[col0 + 16 + n] : 0.0f;
    #pragma unroll
    for (int r = 0; r < 8; ++r) {
        const int rr = row0 + r + 8 * half;
        if (rr < N) {
            float v0 = c0[r] + bias0;
            float v1 = c1[r] + bias1;
            if (do_relu) { v0 = fmaxf(v0, 0.0f); v1 = fmaxf(v1, 0.0f); }
            float* o = out + (size_t)rr * ldo + col_off;
            o[col0 + n]      = v0;
            o[col0 + 16 + n] = v1;
        }
    }
}

// ---------------------------------------------------------------------------
// Degree / normalization helpers
// ---------------------------------------------------------------------------
__global__ void deg_init(float* __restrict__ deg, int N) {
    int i = blockIdx.x * blockDim.x + threadIdx.x;
    if (i < N) deg[i] = 1.0f;                       // self-loop
}

__global__ void deg_count(const int* __restrict__ ei, long long E,
                          float* __restrict__ deg) {
    long long e = (long long)blockIdx.x * blockDim.x + threadIdx.x;
    if (e < E) atomicAdd(&deg[ei[E + e]], 1.0f);    // dst row is second half
}

__global__ void make_dinv(float* __restrict__ deg, int N) {
    int i = blockIdx.x * blockDim.x + threadIdx.x;
    if (i < N) deg[i] = rsqrtf(deg[i]);
}

// ---------------------------------------------------------------------------
// GCN aggregation
// ---------------------------------------------------------------------------
__global__ void gcn_selfloop(const float* __restrict__ H,
                             const float* __restrict__ dinv,
                             float* __restrict__ out, int N) {
    int i = blockIdx.x * blockDim.x + threadIdx.x;  // over N*128
    if (i < N * 128) {
        const float d = dinv[i >> 7];
        out[i] = H[i] * d * d;                      // norm = dinv[i]*dinv[i]
    }
}

__global__ void gcn_scatter(const int* __restrict__ ei, long long E,
                            const float* __restrict__ dinv,
                            const float* __restrict__ H,
                            float* __restrict__ out) {
    long long t = (long long)blockIdx.x * blockDim.x + threadIdx.x;
    long long e = t >> 5;
    int lane = (int)(t & 31);
    if (e < E) {
        const int s = ei[e];
        const int d = ei[E + e];
        const float norm = dinv[s] * dinv[d];
        const float4 v = ((const float4*)(H + (size_t)s * 128))[lane];
        float* o = out + (size_t)d * 128 + lane * 4;
        atomicAdd(o + 0, v.x * norm);
        atomicAdd(o + 1, v.y * norm);
        atomicAdd(o + 2, v.z * norm);
        atomicAdd(o + 3, v.w * norm);
    }
}

// out[node*ldo + col_off + f] = relu(in[node*128 + f] + bias[f])
__global__ void bias_relu_copy(const float* __restrict__ in,
                               const float* __restrict__ bias,
                               float* __restrict__ out,
                               int ldo, int col_off, int N) {
    int i = blockIdx.x * blockDim.x + threadIdx.x;
    if (i < N * 128) {
        const int node = i >> 7, f = i & 127;
        const float v = fmaxf(in[i] + bias[f], 0.0f);
        out[(size_t)node * ldo + col_off + f] = v;
    }
}

// ---------------------------------------------------------------------------
// Final head: sigmoid(X[N,128] @ w[128] + b). One wave per row.
// ---------------------------------------------------------------------------
__global__ void out_sigmoid(const float* __restrict__ X,
                            const float* __restrict__ w,
                            const float* __restrict__ b,
                            float* __restrict__ out, int N) {
    int row  = blockIdx.x * (blockDim.x >> 5) + (threadIdx.x >> 5);
    int lane = threadIdx.x & 31;
    if (row < N) {
        const float4 xv = ((const float4*)(X + (size_t)row * 128))[lane];
        const float4 wv = ((const float4*)w)[lane];
        float s = xv.x * wv.x + xv.y * wv.y + xv.z * wv.z + xv.w * wv.w;
        #pragma unroll
        for (int off = 16; off > 0; off >>= 1) s += __shfl_down(s, off, 32);
        if (lane == 0) out[row] = 1.0f / (1.0f + expf(-(s + b[0])));
    }
}

// ---------------------------------------------------------------------------
extern "C" void kernel_launch(void* const* d_in, const int* in_sizes, int n_in,
                              void* d_out, int out_size, void* d_ws, size_t ws_size,
                              hipStream_t stream) {
    const float* metadata_a = (const float*)d_in[0];
    const float* metadata_b = (const float*)d_in[1];
    const float* x_a        = (const float*)d_in[2];
    const float* x_b        = (const float*)d_in[3];
    const int*   ei_a       = (const int*)d_in[4];
    const int*   ei_b       = (const int*)d_in[5];
    const float* fc1_W  = (const float*)d_in[6];
    const float* fc1_b  = (const float*)d_in[7];
    const float* fc2_W  = (const float*)d_in[8];
    const float* fc2_b  = (const float*)d_in[9];
    const float* gcn1_W = (const float*)d_in[10];
    const float* gcn1_b = (const float*)d_in[11];
    const float* gcn2_W = (const float*)d_in[12];
    const float* gcn2_b = (const float*)d_in[13];
    const float* fcc_W  = (const float*)d_in[14];
    const float* fcc_b  = (const float*)d_in[15];
    const float* outW   = (const float*)d_in[16];
    const float* outb   = (const float*)d_in[17];
    float* outp = (float*)d_out;

    const int N = in_sizes[0] / 512;
    const long long E = in_sizes[4] / 2;

    // Workspace layout (floats): 897*N total (~180 MB for N=50k)
    float* ws       = (float*)d_ws;
    float* combined = ws;                       // [N,512]: ma|mb|ga|gb
    float* tmpH     = ws + (size_t)N * 512;     // [N,128]
    float* tmpT     = ws + (size_t)N * 640;     // [N,128]
    float* fccout   = ws + (size_t)N * 768;     // [N,128]
    float* dinv     = ws + (size_t)N * 896;     // [N]

    const dim3 gemm_grid((N + 15) / 16), gemm_blk(128);
    const int ew_blocks  = (N * 128 + 255) / 256;
    const int n_blocks   = (N + 255) / 256;
    const int e_blocks   = (int)((E + 255) / 256);
    const int sc_blocks  = (int)((E * 32 + 255) / 256);

    // ---- MLP branches (metadata_a / metadata_b) ----
    gemm_bias_act_wmma<<<gemm_grid, gemm_blk, 0, stream>>>(
        metadata_a, fc1_W, fc1_b, tmpH, N, 512, 128, 0, 1);
    gemm_bias_act_wmma<<<gemm_grid, gemm_blk, 0, stream>>>(
        tmpH, fc2_W, fc2_b, combined, N, 128, 512, 0, 1);
    gemm_bias_act_wmma<<<gemm_grid, gemm_blk, 0, stream>>>(
        metadata_b, fc1_W, fc1_b, tmpH, N, 512, 128, 0, 1);
    gemm_bias_act_wmma<<<gemm_grid, gemm_blk, 0, stream>>>(
        tmpH, fc2_W, fc2_b, combined, N, 128, 512, 128, 1);

    // ---- GCN branches ----
    for (int g = 0; g < 2; ++g) {
        const int*   ei = g ? ei_b : ei_a;
        const float* x  = g ? x_b  : x_a;
        const int colo  = g ? 384  : 256;

        deg_init<<<n_blocks, 256, 0, stream>>>(dinv, N);
        deg_count<<<e_blocks, 256, 0, stream>>>(ei, E, dinv);
        make_dinv<<<n_blocks, 256, 0, stream>>>(dinv, N);

        // layer 1: H = x @ W1 ; aggregate ; +b1 ; relu  (in-place in tmpT)
        gemm_bias_act_wmma<<<gemm_grid, gemm_blk, 0, stream>>>(
            x, gcn1_W, nullptr, tmpH, N, 256, 128, 0, 0);
        gcn_selfloop<<<ew_blocks, 256, 0, stream>>>(tmpH, dinv, tmpT, N);
        gcn_scatter<<<sc_blocks, 256, 0, stream>>>(ei, E, dinv, tmpH, tmpT);
        bias_relu_copy<<<ew_blocks, 256, 0, stream>>>(tmpT, gcn1_b, tmpT, 128, 0, N);

        // layer 2: H = tmpT @ W2 ; aggregate ; +b2 ; relu -> combined slice
        gemm_bias_act_wmma<<<gemm_grid, gemm_blk, 0, stream>>>(
            tmpT, gcn2_W, nullptr, tmpH, N, 128, 128, 0, 0);
        gcn_selfloop<<<ew_blocks, 256, 0, stream>>>(tmpH, dinv, tmpT, N);
        gcn_scatter<<<sc_blocks, 256, 0, stream>>>(ei, E, dinv, tmpH, tmpT);
        bias_relu_copy<<<ew_blocks, 256, 0, stream>>>(tmpT, gcn2_b, combined, 512, colo, N);
    }

    // ---- Combine + head ----
    gemm_bias_act_wmma<<<gemm_grid, gemm_blk, 0, stream>>>(
        combined, fcc_W, fcc_b, fccout, N, 512, 128, 0, 1);
    out_sigmoid<<<(N + 7) / 8, 256, 0, stream>>>(fccout, outW, outb, outp, N);
}